// TemporalGNN_28784870818367
// MI455X (gfx1250) — compile-verified
//
#include <hip/hip_runtime.h>

typedef __attribute__((ext_vector_type(2))) float v2f;
typedef __attribute__((ext_vector_type(8))) float v8f;

#define DD 128
#define NBASES 10
#define NRELS 200
#define NNODES 50000
#define NEDGES 800000
#define LPAD 4                     // LDS row padding (floats) to avoid bank conflicts
#define SEGPAD 64                  // relation segments padded to multiples of 64 edges
#define PADMAX (NEDGES + NRELS*SEGPAD) // 812800: worst-case padded permutation length
#define NEBLK (PADMAX / 64)        // 12700 edge blocks (64 edges each)
#define ROWB ((NNODES + 63) / 64)  // 782 row blocks (64 rows each)

// ---------------- small utility kernels ----------------

__global__ void zero_f32(float* __restrict__ p, int n) {
    int i = blockIdx.x * 256 + threadIdx.x;
    if (i < n) p[i] = 0.0f;
}

__global__ void fill_i32(int* __restrict__ p, int n, int val) {
    int i = blockIdx.x * 256 + threadIdx.x;
    if (i < n) p[i] = val;
}

__global__ void hist_kernel(const int* __restrict__ rel, int* __restrict__ counts) {
    int e = blockIdx.x * 256 + threadIdx.x;
    if (e < NEDGES) atomicAdd(&counts[rel[e]], 1);
}

// single thread: 64-padded exclusive prefix over 200 bins
__global__ void offsets_kernel(const int* __restrict__ counts, int* __restrict__ cursor) {
    if (threadIdx.x == 0 && blockIdx.x == 0) {
        int off = 0;
        for (int r = 0; r < NRELS; ++r) {
            cursor[r] = off;
            off += ((counts[r] + SEGPAD - 1) / SEGPAD) * SEGPAD;
        }
    }
}

__global__ void scatter_kernel(const int* __restrict__ rel, int* __restrict__ cursor,
                               int* __restrict__ perm) {
    int e = blockIdx.x * 256 + threadIdx.x;
    if (e < NEDGES) {
        int r = rel[e];
        int pos = atomicAdd(&cursor[r], 1);
        perm[pos] = e;
    }
}

// W[r][k][d] = sum_b comp[r][b] * bases[b][k][d]   (one layer)
__global__ void relweight_kernel(const float* __restrict__ comp, const float* __restrict__ bases,
                                 float* __restrict__ W) {
    long i = (long)blockIdx.x * 256 + threadIdx.x;   // over NRELS*128*128
    if (i >= (long)NRELS * DD * DD) return;
    int r  = (int)(i / (DD * DD));
    int kd = (int)(i % (DD * DD));
    float s = 0.0f;
    #pragma unroll
    for (int b = 0; b < NBASES; ++b)
        s += comp[r * NBASES + b] * bases[(long)b * DD * DD + kd];
    W[i] = s;
}

// ---------------- WMMA fp32 mainloop, 4-way row-blocked ----------------
// One wave computes FOUR 16x16 fp32 tiles (64 rows x 16 cols): each B fragment
// (2 global loads, L2-hot) feeds 4 independent v_wmma_f32_16x16x4_f32 chains.

__device__ __forceinline__ void wmma_mainloop4(const float* As, int rowstride, int K,
                                               const float* __restrict__ B, int colBase,
                                               int lane, v8f acc[4]) {
    const int mrow = lane & 15;
    const int koff = (lane >> 4) * 2;
    const int col  = colBase + (lane & 15);
    for (int k0 = 0; k0 < K; k0 += 4) {
        const int kk = k0 + koff;
        v2f b;
        b.x = B[kk * DD + col];
        b.y = B[(kk + 1) * DD + col];
        #pragma unroll
        for (int rt = 0; rt < 4; ++rt) {
            v2f a;
            a.x = As[(rt * 16 + mrow) * rowstride + kk];
            a.y = As[(rt * 16 + mrow) * rowstride + kk + 1];
            acc[rt] = __builtin_amdgcn_wmma_f32_16x16x4_f32(false, a, false, b,
                                                            (short)0, acc[rt], false, false);
        }
    }
}

// ---------------- fusion: h = relu(concat(ent,time) @ Wf + bf) ----------------
// grid: ROWB blocks of 256 (8 waves; wave w -> 16-col tile w; 64 rows/block)

__global__ __launch_bounds__(256)
void fusion_kernel(const int* __restrict__ eid, const int* __restrict__ tim,
                   const float* __restrict__ ent, const float* __restrict__ ttab,
                   const float* __restrict__ Wf /*256x128*/, const float* __restrict__ bf,
                   float* __restrict__ h) {
    __shared__ float As[64 * (2 * DD + LPAD)];
    const int rowBase = blockIdx.x * 64;
    const int t = threadIdx.x;
    for (int i = t; i < 64 * 2 * DD; i += 256) {
        int m = i >> 8, k = i & 255;
        int node = rowBase + m;
        float v = 0.0f;
        if (node < NNODES)
            v = (k < DD) ? ent[(long)eid[node] * DD + k]
                         : ttab[(long)tim[node] * DD + (k - DD)];
        As[m * (2 * DD + LPAD) + k] = v;
    }
    __syncthreads();
    const int lane = t & 31, wave = t >> 5;
    const int colBase = wave * 16;
    v8f acc[4] = {};
    wmma_mainloop4(As, 2 * DD + LPAD, 2 * DD, Wf, colBase, lane, acc);
    const int col = colBase + (lane & 15);
    const float bias = bf[col];
    #pragma unroll
    for (int rt = 0; rt < 4; ++rt) {
        #pragma unroll
        for (int v = 0; v < 8; ++v) {
            int r = rowBase + rt * 16 + v + ((lane >> 4) << 3);
            if (r < NNODES) {
                float val = acc[rt][v] + bias;
                h[(long)r * DD + col] = val > 0.0f ? val : 0.0f;
            }
        }
    }
}

// ---------------- edge messages: agg[dst] += h[src] @ W[rel] (tiled WMMA) ----------------
// grid: NEBLK blocks of 256; every 64-edge block is relation-uniform (padded with -1)

__global__ __launch_bounds__(256)
void edgemsg_kernel(const int* __restrict__ perm, const int* __restrict__ src,
                    const int* __restrict__ dst, const int* __restrict__ rel,
                    const float* __restrict__ h, const float* __restrict__ W,
                    float* __restrict__ agg) {
    __shared__ float As[64 * (DD + LPAD)];
    __shared__ int eIds[64];
    __shared__ int sRel;
    const int t = threadIdx.x;
    if (t < 64) eIds[t] = perm[blockIdx.x * 64 + t];
    __syncthreads();
    if (t == 0) {
        int r = -1;
        for (int m = 0; m < 64; ++m)
            if (eIds[m] >= 0) { r = rel[eIds[m]]; break; }
        sRel = r;
    }
    __syncthreads();
    if (sRel < 0) return;                 // fully-padded block: uniform exit
    for (int i = t; i < 64 * DD; i += 256) {
        int m = i >> 7, k = i & 127;
        int e = eIds[m];
        As[m * (DD + LPAD) + k] = (e >= 0) ? h[(long)src[e] * DD + k] : 0.0f;
    }
    __syncthreads();
    const int lane = t & 31, wave = t >> 5;
    const int colBase = wave * 16;
    const float* B = W + (long)sRel * DD * DD;
    v8f acc[4] = {};
    wmma_mainloop4(As, DD + LPAD, DD, B, colBase, lane, acc);
    const int col = colBase + (lane & 15);
    #pragma unroll
    for (int rt = 0; rt < 4; ++rt) {
        #pragma unroll
        for (int v = 0; v < 8; ++v) {
            int m = rt * 16 + v + ((lane >> 4) << 3);
            int e = eIds[m];
            if (e >= 0) atomicAdd(&agg[(long)dst[e] * DD + col], acc[rt][v]);
        }
    }
}

// ---------------- combine: hout = relu(h @ Wself + agg + b) ----------------

__global__ __launch_bounds__(256)
void selfcombine_kernel(const float* __restrict__ h, const float* __restrict__ Wself,
                        const float* __restrict__ agg, const float* __restrict__ bias,
                        float* __restrict__ hout) {
    __shared__ float As[64 * (DD + LPAD)];
    const int rowBase = blockIdx.x * 64;
    const int t = threadIdx.x;
    for (int i = t; i < 64 * DD; i += 256) {
        int m = i >> 7, k = i & 127;
        int node = rowBase + m;
        As[m * (DD + LPAD) + k] = (node < NNODES) ? h[(long)node * DD + k] : 0.0f;
    }
    __syncthreads();
    const int lane = t & 31, wave = t >> 5;
    const int colBase = wave * 16;
    v8f acc[4] = {};
    wmma_mainloop4(As, DD + LPAD, DD, Wself, colBase, lane, acc);
    const int col = colBase + (lane & 15);
    const float b = bias[col];
    #pragma unroll
    for (int rt = 0; rt < 4; ++rt) {
        #pragma unroll
        for (int v = 0; v < 8; ++v) {
            int r = rowBase + rt * 16 + v + ((lane >> 4) << 3);
            if (r < NNODES) {
                float val = acc[rt][v] + agg[(long)r * DD + col] + b;
                hout[(long)r * DD + col] = val > 0.0f ? val : 0.0f;
            }
        }
    }
}

// ---------------- host launch ----------------

extern "C" void kernel_launch(void* const* d_in, const int* in_sizes, int n_in,
                              void* d_out, int out_size, void* d_ws, size_t ws_size,
                              hipStream_t stream) {
    const int*   entity_ids = (const int*)d_in[0];
    const int*   time_ids   = (const int*)d_in[1];
    const int*   rel_ids    = (const int*)d_in[2];
    const int*   src        = (const int*)d_in[3];
    const int*   dst        = (const int*)d_in[4];
    const float* ent_table  = (const float*)d_in[5];
    const float* time_table = (const float*)d_in[6];
    const float* fusion_w   = (const float*)d_in[7];
    const float* fusion_b   = (const float*)d_in[8];
    const float* comp       = (const float*)d_in[9];   // [3,200,10]
    const float* bases      = (const float*)d_in[10];  // [3,10,128,128]
    const float* w_self     = (const float*)d_in[11];  // [3,128,128]
    const float* layer_b    = (const float*)d_in[12];  // [3,128]

    // workspace carving (floats then ints); total ~93 MB
    float* hA   = (float*)d_ws;                         // 50000*128
    float* hB   = hA + (long)NNODES * DD;               // 50000*128
    float* agg  = hB + (long)NNODES * DD;               // 50000*128
    float* W    = agg + (long)NNODES * DD;              // 200*128*128
    int*   counts = (int*)(W + (long)NRELS * DD * DD);  // 256
    int*   cursor = counts + 256;                       // 256
    int*   perm   = cursor + 256;                       // PADMAX

    const int EB = (NEDGES + 255) / 256;                 // 3125
    const int PB = (PADMAX + 255) / 256;                 // perm fill blocks
    const int WB = ((long)NRELS * DD * DD + 255) / 256;  // 12800

    // --- relation counting sort (rel_ids is the same for all layers) ---
    fill_i32<<<1, 256, 0, stream>>>(counts, NRELS, 0);
    hist_kernel<<<EB, 256, 0, stream>>>(rel_ids, counts);
    offsets_kernel<<<1, 32, 0, stream>>>(counts, cursor);
    fill_i32<<<PB, 256, 0, stream>>>(perm, PADMAX, -1);
    scatter_kernel<<<EB, 256, 0, stream>>>(rel_ids, cursor, perm);

    // --- embedding fusion ---
    fusion_kernel<<<ROWB, 256, 0, stream>>>(entity_ids, time_ids, ent_table, time_table,
                                            fusion_w, fusion_b, hA);

    float* bufs[3] = { hA, hB, hA };
    for (int l = 0; l < 3; ++l) {
        const float* hin  = bufs[l];
        float*       hout = (l == 2) ? (float*)d_out : bufs[l + 1];
        relweight_kernel<<<WB, 256, 0, stream>>>(comp + (long)l * NRELS * NBASES,
                                                 bases + (long)l * NBASES * DD * DD, W);
        zero_f32<<<(NNODES * DD + 255) / 256, 256, 0, stream>>>(agg, NNODES * DD);
        edgemsg_kernel<<<NEBLK, 256, 0, stream>>>(perm, src, dst, rel_ids, hin, W, agg);
        selfcombine_kernel<<<ROWB, 256, 0, stream>>>(hin, w_self + (long)l * DD * DD,
                                                     agg, layer_b + (long)l * DD, hout);
    }
}